// JointNet_25082609008778
// MI455X (gfx1250) — compile-verified
//
#include <hip/hip_runtime.h>

// Problem dims (fixed by reference)
#define B_ 4
#define T_ 512
#define U_ 100
#define D_ 512
#define C_ 1024

typedef __attribute__((ext_vector_type(2))) float v2f;
typedef __attribute__((ext_vector_type(4))) float v4f;
typedef __attribute__((ext_vector_type(8))) float v8f;

// ---------------------------------------------------------------------------
// Phase 1: proj[m, c] = sum_d X[m, d] * W[c, woff + d]   (X @ W_sub^T)
// One wave computes a 16(M) x 64(N) tile via 4 accumulators of
// V_WMMA_F32_16X16X4_F32, looping K in steps of 4 (full fp32 precision).
//
// f32 16x4 A fragment layout (ISA 7.12.2): lanes 0-15 hold M=lane, K={0,1};
// lanes 16-31 hold M=lane-16, K={2,3}; VGPR i = K offset i. So each lane's
// fragment is a contiguous 2-float (8B-aligned) load. B (4x16) is symmetric
// with N striped across lanes, which for B = W^T means each lane also reads
// 2 contiguous floats along a row of W.
// ---------------------------------------------------------------------------
__global__ __launch_bounds__(256) void joint_gemm_wmma(
    const float* __restrict__ X, int M,
    const float* __restrict__ W, int woff,
    float* __restrict__ out)
{
    const int wave = threadIdx.x >> 5;
    const int lane = threadIdx.x & 31;
    const int lo   = lane & 15;
    const int hi   = lane >> 4;

    const int tilesN   = C_ / 64;                 // 16
    const int tileId   = blockIdx.x * 8 + wave;   // 8 waves per block
    const int numTiles = (M >> 4) * tilesN;
    if (tileId >= numTiles) return;               // wave-uniform guard (EXEC stays all-1s)

    const int m0 = (tileId / tilesN) * 16;
    const int n0 = (tileId % tilesN) * 64;

    const float* Arow = X + (size_t)(m0 + lo) * D_ + 2 * hi;
    const float* B0p  = W + (size_t)(n0 +  0 + lo) * (2 * D_) + woff + 2 * hi;
    const float* B1p  = W + (size_t)(n0 + 16 + lo) * (2 * D_) + woff + 2 * hi;
    const float* B2p  = W + (size_t)(n0 + 32 + lo) * (2 * D_) + woff + 2 * hi;
    const float* B3p  = W + (size_t)(n0 + 48 + lo) * (2 * D_) + woff + 2 * hi;

    v8f acc0 = {}, acc1 = {}, acc2 = {}, acc3 = {};

#pragma unroll 4
    for (int k = 0; k < D_; k += 4) {
        v2f a  = *(const v2f*)(Arow + k);
        v2f b0 = *(const v2f*)(B0p + k);
        v2f b1 = *(const v2f*)(B1p + k);
        v2f b2 = *(const v2f*)(B2p + k);
        v2f b3 = *(const v2f*)(B3p + k);
        // (neg_a, A, neg_b, B, c_mod, C, reuse_a, reuse_b)
        acc0 = __builtin_amdgcn_wmma_f32_16x16x4_f32(false, a, false, b0, (short)0, acc0, false, false);
        acc1 = __builtin_amdgcn_wmma_f32_16x16x4_f32(false, a, false, b1, (short)0, acc1, false, false);
        acc2 = __builtin_amdgcn_wmma_f32_16x16x4_f32(false, a, false, b2, (short)0, acc2, false, false);
        acc3 = __builtin_amdgcn_wmma_f32_16x16x4_f32(false, a, false, b3, (short)0, acc3, false, false);
    }

    // f32 16x16 C/D layout: VGPR v, lanes 0-15 -> (M = v,   N = lane),
    //                               lanes 16-31 -> (M = v+8, N = lane-16)
    float* orow = out + (size_t)(m0 + 8 * hi) * C_ + n0 + lo;
#pragma unroll
    for (int v = 0; v < 8; ++v) {
        float* p = orow + (size_t)v * C_;
        p[0]  = acc0[v];
        p[16] = acc1[v];
        p[32] = acc2[v];
        p[48] = acc3[v];
    }
}

// ---------------------------------------------------------------------------
// Phase 2: out[b,t,u,c] = enc_proj[b,t,c] + dec_proj[b,u,c]
// One block per (b,t). Thread owns one float4 of C (C/4 = 256 = blockDim).
// enc row cached in registers; dec_proj (1.6 MB) lives in L2; 839 MB of
// write-once output goes out via nontemporal b128 stores.
// ---------------------------------------------------------------------------
__global__ __launch_bounds__(256) void joint_bcast_add(
    const float* __restrict__ encp,
    const float* __restrict__ decp,
    float* __restrict__ out)
{
    const int C4 = C_ / 4;          // 256
    const int bt = blockIdx.x;      // b*T + t
    const int b  = bt / T_;
    const int c4 = threadIdx.x;

    const v4f* e4 = (const v4f*)encp;
    const v4f* d4 = (const v4f*)decp;
    v4f*       o4 = (v4f*)out;

    v4f e = e4[(size_t)bt * C4 + c4];
    const v4f* drow = d4 + (size_t)b * U_ * C4 + c4;
    v4f*       orow = o4 + (size_t)bt * U_ * C4 + c4;

#pragma unroll 4
    for (int u = 0; u < U_; ++u) {
        v4f d = drow[(size_t)u * C4];
        v4f r = e + d;
        __builtin_nontemporal_store(r, orow + (size_t)u * C4);
    }
}

// ---------------------------------------------------------------------------
extern "C" void kernel_launch(void* const* d_in, const int* in_sizes, int n_in,
                              void* d_out, int out_size, void* d_ws, size_t ws_size,
                              hipStream_t stream) {
    const float* enc = (const float*)d_in[0];   // (B, T, D)
    const float* dec = (const float*)d_in[1];   // (B, U, D)
    const float* W   = (const float*)d_in[2];   // (C, 2D)
    float* out = (float*)d_out;                 // (B, T, U, C)

    float* enc_proj = (float*)d_ws;                          // (B*T, C) = 8 MB
    float* dec_proj = enc_proj + (size_t)B_ * T_ * C_;       // (B*U, C) = 1.6 MB

    // enc GEMM: (2048/16)*(1024/64) = 2048 tiles, 8 waves/block -> 256 blocks
    joint_gemm_wmma<<<(B_ * T_ / 16) * (C_ / 64) / 8, 256, 0, stream>>>(
        enc, B_ * T_, W, 0, enc_proj);
    // dec GEMM: (400/16)*16 = 400 tiles -> 50 blocks
    joint_gemm_wmma<<<(B_ * U_ / 16) * (C_ / 64) / 8, 256, 0, stream>>>(
        dec, B_ * U_, W, D_, dec_proj);

    // Broadcast add: one block per (b,t)
    joint_bcast_add<<<B_ * T_, 256, 0, stream>>>(enc_proj, dec_proj, out);
}